// WTA_46875273068758
// MI455X (gfx1250) — compile-verified
//
#include <hip/hip_runtime.h>
#include <stdint.h>

// Problem constants (from reference): x = [16, 64, 128, 128, 2] f32
#define N_SPATIAL  16384                    // H*W (sort axis length)
#define CHANNELS   2
#define TILE_ELEMS (N_SPATIAL * CHANNELS)   // 32768 elems per (b,t) tile
#define TILE_BYTES (TILE_ELEMS * 4)         // 131072 B = 128 KB
// threshold_index = int32(f32(16384) - f32(16384)*f32(0.05)) = 15564 -> keep 820
#define KEEP 820u
#define THREADS 1024

// Keys of +0.0f / -0.0f under the monotone transform (for the v != 0 test)
#define KEY_POS0 0x80000000u
#define KEY_NEG0 0x7FFFFFFFu
#define ONE_BITS 0x3F800000u                // bit pattern of 1.0f

// Monotone map on raw float bits: key order == float ascending order
__device__ __forceinline__ uint32_t bkey(uint32_t u) {
    return (u & 0x80000000u) ? ~u : (u | 0x80000000u);
}

__device__ __forceinline__ void wait_async0() {
#if __has_builtin(__builtin_amdgcn_s_wait_asynccnt)
    __builtin_amdgcn_s_wait_asynccnt(0);
#else
    asm volatile("s_wait_asynccnt 0x0" ::: "memory");
#endif
}

__global__ __launch_bounds__(THREADS, 1)
void wta_topk_mask_kernel(const float* __restrict__ x,
                          float* __restrict__ out,
                          int nTiles) {
    // Tile staged as raw u32; pass 0 converts in place to sortable keys,
    // mask pass converts in place to 0.0f/1.0f bit patterns.
    __shared__ __align__(16) uint32_t s_data[TILE_ELEMS];   // 128 KB
    __shared__ uint32_t s_hist[2][256];
    __shared__ uint32_t s_prefix[2];   // cutoff key bits found so far
    __shared__ uint32_t s_rem[2];      // elements still needed from current prefix set
    __shared__ uint32_t s_m[2];        // # elements exactly equal to cutoff key
    __shared__ uint32_t s_nstar[2];    // spatial-index cutoff for ties
    __shared__ uint32_t s_cnt;

    const uint32_t tid = threadIdx.x;
    const uint32_t wg  = blockIdx.x;
    if ((int)wg >= nTiles) return;     // uniform per block

    const float* tileIn  = x   + (size_t)wg * TILE_ELEMS;
    float*       tileOut = out + (size_t)wg * TILE_ELEMS;

    // ---------------- async global -> LDS (whole 128 KB tile, NT) ------------
    const uint32_t ldsBase = (uint32_t)(uintptr_t)(void*)&s_data[0];
#pragma unroll
    for (int k = 0; k < 8; ++k) {
        uint32_t off     = tid * 16u + (uint32_t)k * 16384u;   // byte offset
        uint32_t ldsAddr = ldsBase + off;
        asm volatile("global_load_async_to_lds_b128 %0, %1, %2 th:TH_LOAD_NT"
                     :: "v"(ldsAddr), "v"(off), "s"(tileIn)
                     : "memory");
    }
    if (tid < 2) {
        s_prefix[tid] = 0u;
        s_rem[tid]    = KEEP;
        s_m[tid]      = 0u;
        s_nstar[tid]  = 0u;
    }
    if (tid < 512) {                   // zero histograms once; scan waves re-zero
        s_hist[tid >> 8][tid & 255u] = 0u;
    }
    wait_async0();          // this wave's async loads landed in LDS
    __syncthreads();        // all waves' loads landed, control vars ready

    // Thread -> (channel, strided chunk) mapping. 512 threads per channel,
    // each owns n = lane + 512*j (j=0..31). LDS word = n*2+c -> 32 distinct
    // banks per wave (bank = (2*lane + c) & 63).
    const uint32_t cLane = tid >> 9;         // 0 or 1
    const uint32_t lane  = tid & 511u;

    // ---------------- 4-pass MSB-first radix select (8-bit digits) -----------
#pragma unroll
    for (int p = 0; p < 4; ++p) {
        const int shHi = 32 - 8 * p;         // bits decided so far
        const int shLo = 24 - 8 * p;         // current digit position
        {
            const uint32_t pref = s_prefix[cLane];
#pragma unroll 4
            for (int j = 0; j < 32; ++j) {
                uint32_t n   = lane + 512u * (uint32_t)j;
                uint32_t idx = n * 2u + cLane;
                if (p == 0) {
                    uint32_t key = bkey(s_data[idx]);
                    s_data[idx] = key;                     // in-place key transform
                    atomicAdd(&s_hist[cLane][key >> 24], 1u);
                } else {
                    uint32_t key = s_data[idx];
                    if (((key ^ pref) >> shHi) == 0u)
                        atomicAdd(&s_hist[cLane][(key >> shLo) & 255u], 1u);
                }
            }
        }
        __syncthreads();

        // Single-wave suffix scan + boundary detect per channel (no block
        // barriers inside): wave 0 -> channel 0, wave 1 -> channel 1.
        // Lane L owns bins [8L, 8L+7].
        if (tid < 64) {
            const uint32_t c = tid >> 5;
            const uint32_t L = tid & 31u;
            uint32_t h[8];
            uint32_t gsum = 0u;
#pragma unroll
            for (int j = 0; j < 8; ++j) { h[j] = s_hist[c][L * 8u + j]; gsum += h[j]; }
            // wave32 inclusive suffix scan of group sums (ds_bpermute shuffles)
            uint32_t incl = gsum;
#pragma unroll
            for (int off = 1; off < 32; off <<= 1) {
                uint32_t add = __shfl_down(incl, off, 32);
                if (L + (uint32_t)off >= 32u) add = 0u;
                incl += add;
            }
            const uint32_t above = incl - gsum;   // suffix of groups strictly above
            const uint32_t rem   = s_rem[c];
            // Walk own 8 bins high->low: S(bin) = run + h[bin]; boundary bin b
            // satisfies S(b) >= rem > S(b+1). Unique across the whole wave.
            uint32_t run = above;
            int      bfound = -1;
            uint32_t nxt = 0u, me = 0u;
#pragma unroll
            for (int j = 7; j >= 0; --j) {
                uint32_t s = run + h[j];
                if (s >= rem && run < rem) { bfound = j; nxt = run; me = s; }
                run = s;
            }
            if (bfound >= 0) {
                uint32_t d = L * 8u + (uint32_t)bfound;
                s_prefix[c] |= d << shLo;
                s_rem[c]     = rem - nxt;          // bins > d are all kept
                if (p == 3) s_m[c] = me - nxt;     // multiplicity of cutoff key
            }
            // re-zero bins for the next pass
#pragma unroll
            for (int j = 0; j < 8; ++j) s_hist[c][L * 8u + j] = 0u;
        }
        __syncthreads();
    }

    // ---------------- tie resolution (rank tie-break = larger index wins) ----
    // Keep the last r (by spatial index n) of the m elements equal to K:
    // n* = smallest n0 with |{n >= n0 : key==K}| <= r. m==r => n* = 0.
    for (int c = 0; c < 2; ++c) {
        uint32_t m = s_m[c], r = s_rem[c];           // block-uniform values
        if (m > r) {
            const uint32_t K = s_prefix[c];
            uint32_t lo = 0u, hi = N_SPATIAL;
            for (int it = 0; it < 15; ++it) {        // fixed trip count: uniform barriers
                uint32_t mid = (lo + hi) >> 1;
                if (tid == 0) s_cnt = 0u;
                __syncthreads();
                if (cLane == (uint32_t)c) {
                    uint32_t local = 0u;
#pragma unroll 4
                    for (int j = 0; j < 32; ++j) {
                        uint32_t n = lane + 512u * (uint32_t)j;
                        if (n >= mid && s_data[n * 2u + c] == K) local++;
                    }
                    if (local) atomicAdd(&s_cnt, local);
                }
                __syncthreads();
                uint32_t f = s_cnt;
                __syncthreads();
                if (lo < hi) { if (f <= r) hi = mid; else lo = mid + 1u; }
            }
            if (tid == 0) s_nstar[c] = lo;
            __syncthreads();
        }
    }

    // ---------------- mask pass: overwrite keys in LDS with 0/1 bits ---------
    {
        const uint32_t K  = s_prefix[cLane];
        const uint32_t ns = s_nstar[cLane];
#pragma unroll 4
        for (int j = 0; j < 32; ++j) {
            uint32_t n   = lane + 512u * (uint32_t)j;
            uint32_t idx = n * 2u + cLane;
            uint32_t key = s_data[idx];
            bool keep = (key > K) || ((key == K) && (n >= ns));
            bool nz   = (key != KEY_POS0) && (key != KEY_NEG0);   // original v != 0
            s_data[idx] = (keep && nz) ? ONE_BITS : 0u;
        }
    }
    __syncthreads();

    // ---------------- async LDS -> global (coalesced b128 NT stores) ---------
#pragma unroll
    for (int k = 0; k < 8; ++k) {
        uint32_t off     = tid * 16u + (uint32_t)k * 16384u;
        uint32_t ldsAddr = ldsBase + off;
        asm volatile("global_store_async_from_lds_b128 %0, %1, %2 th:TH_STORE_NT"
                     :: "v"(off), "v"(ldsAddr), "s"(tileOut)
                     : "memory");
    }
    wait_async0();   // (s_endpgm also implicitly waits-idle)
}

extern "C" void kernel_launch(void* const* d_in, const int* in_sizes, int n_in,
                              void* d_out, int out_size, void* d_ws, size_t ws_size,
                              hipStream_t stream) {
    (void)n_in; (void)out_size; (void)d_ws; (void)ws_size;
    const float* x   = (const float*)d_in[0];
    float*       out = (float*)d_out;
    int nTiles = in_sizes[0] / TILE_ELEMS;   // = B*T = 1024
    wta_topk_mask_kernel<<<dim3((unsigned)nTiles), dim3(THREADS), 0, stream>>>(x, out, nTiles);
}